// Model_69767448756495
// MI455X (gfx1250) — compile-verified
//
#include <hip/hip_runtime.h>

// Scatter-add: out = self_tensor; out[index[i,j], j] += updates[i,j]
// N=100000, M=50000, D=256 (fp32). Memory/atomic-bound; no contraction dim,
// so WMMA is not applicable. Optimize the CDNA5 memory path instead:
//  - 128-bit vector loads (global_load_b128) with NT temporal hints on the
//    read-once streams (self_tensor, updates, index) to preserve L2 for out
//  - non-returning hardware fp32 atomics (global_atomic_add_f32, STOREcnt)
//  - 32-bit element offsets so atomics use SADDR+VGPR32 (GVS) addressing
//    with scale_offset instead of 4x 64-bit address chains
//  - flat one-element-per-thread kernels: MLP comes from 100k wave32s in
//    flight, not per-thread loops
// Output (102.4 MB) fits in the 192 MB L2, so all 12.8M atomics are L2 RMWs.

typedef float v4f __attribute__((ext_vector_type(4)));
typedef int   v4i __attribute__((ext_vector_type(4)));

static constexpr int DCOLS = 256;   // reference D; power of two

// One vec4 (4 consecutive columns of one updates row) per thread.
__global__ void __launch_bounds__(256) scatter_add_kernel(const v4f* __restrict__ upd,
                                                          const v4i* __restrict__ idx,
                                                          float* __restrict__ out,
                                                          int n4) {
    const int i = blockIdx.x * blockDim.x + threadIdx.x;
    if (i >= n4) return;

    // Coalesced 128-bit NT loads; j is the fast axis so 4 consecutive flat
    // elements share one updates row (D=256 divisible by 4).
    v4f u = __builtin_nontemporal_load(upd + i);
    v4i x = __builtin_nontemporal_load(idx + i);

    const unsigned c0 = ((unsigned)i * 4u) & (DCOLS - 1);  // starting column

    // 32-bit unsigned element offsets (max 25.6M < 2^31): backend can emit
    // global_atomic_add_f32 with SGPR base + 32-bit VGPR offset (scale_offset)
    // -> one v_lshl_or_b32 of address math per atomic, no 64-bit chains.
    const unsigned o0 = ((unsigned)x.x << 8) | (c0 + 0u);
    const unsigned o1 = ((unsigned)x.y << 8) | (c0 + 1u);
    const unsigned o2 = ((unsigned)x.z << 8) | (c0 + 2u);
    const unsigned o3 = ((unsigned)x.w << 8) | (c0 + 3u);

    // Non-returning hardware fp32 atomics -> global_atomic_add_f32,
    // tracked on STOREcnt only (fire-and-forget, no dependency stall).
    unsafeAtomicAdd(out + o0, u.x);
    unsafeAtomicAdd(out + o1, u.y);
    unsafeAtomicAdd(out + o2, u.z);
    unsafeAtomicAdd(out + o3, u.w);
}

__global__ void __launch_bounds__(256) copy_rows_kernel(const v4f* __restrict__ src,
                                                        v4f* __restrict__ dst,
                                                        int n4) {
    const int i = blockIdx.x * blockDim.x + threadIdx.x;
    if (i < n4) {
        // NT load: self_tensor is read exactly once -> don't let it evict the
        // output working set from L2. Store with default RT so the destination
        // stays L2-resident for the atomic phase.
        v4f v = __builtin_nontemporal_load(src + i);
        dst[i] = v;
    }
}

extern "C" void kernel_launch(void* const* d_in, const int* in_sizes, int n_in,
                              void* d_out, int out_size, void* d_ws, size_t ws_size,
                              hipStream_t stream) {
    // setup_inputs order: self_tensor (N*D f32), index (M*D i32),
    //                     updates (M*D f32), axis (scalar, ignored: always 0)
    const float* self_tensor = (const float*)d_in[0];
    const int*   index       = (const int*)d_in[1];
    const float* updates     = (const float*)d_in[2];
    float*       out         = (float*)d_out;

    const int nd = in_sizes[0];       // N*D = 25,600,000
    const int md = in_sizes[1];       // M*D = 12,800,000

    // Phase 1: out = self_tensor (vectorized, one float4 per thread).
    {
        const int n4 = nd / 4;        // 6,400,000 float4
        const int threads = 256;
        const int blocks  = (n4 + threads - 1) / threads;
        copy_rows_kernel<<<blocks, threads, 0, stream>>>(
            (const v4f*)self_tensor, (v4f*)out, n4);
    }

    // Phase 2: scattered atomic adds, one vec4 per thread.
    {
        const int n4 = md / 4;        // 3,200,000 vec4 elements
        const int threads = 256;      // 8 waves per block (wave32)
        const int blocks  = (n4 + threads - 1) / threads;   // 12,500
        scatter_add_kernel<<<blocks, threads, 0, stream>>>(
            (const v4f*)updates, (const v4i*)index, out, n4);
    }
}